// ReGULayer_32667521253541
// MI455X (gfx1250) — compile-verified
//
#include <hip/hip_runtime.h>
#include <hip/hip_bf16.h>
#include <math.h>

// ---------------------------------------------------------------------------
// ReGU layer for MI455X (gfx1250): fp32 WMMA (v_wmma_f32_16x16x4_f32) GEMMs.
//   T=512, B=32, H=1024.
//   Phase 1: xg = inputs @ w_ih^T + b_ih            (big parallel WMMA GEMM)
//   Phase 2: 512 sequential step kernels:
//            gates = sigmoid(xg_t + c @ w_hh^T + b_hh)
//            c'    = (1-f)*c + f*tanh(x)
//            h     = (1-o)*x + o*tanh(c')
//   c is double-buffered across kernel launches (stream order = sync).
// ---------------------------------------------------------------------------

typedef float v2f __attribute__((ext_vector_type(2)));
typedef float v8f __attribute__((ext_vector_type(8)));

#define TT 512
#define BB 32
#define HH 1024
#define G2 (2 * HH)   // 2048

__device__ __forceinline__ v8f wmma_f32(v2f a, v2f b, v8f c) {
    // D = A(16x4) * B(4x16) + C(16x16), fp32, wave32.
    return __builtin_amdgcn_wmma_f32_16x16x4_f32(
        /*neg_a=*/false, a, /*neg_b=*/false, b,
        /*c_mod=*/(short)0, c, /*reuse_a=*/false, /*reuse_b=*/false);
}

__device__ __forceinline__ float sigmoidf_(float x) {
    return 1.0f / (1.0f + expf(-x));
}

// ---------------------------------------------------------------------------
// Phase 1: XG[M=T*B, 2H] = X[M, H] @ Wih^T ([2H, H] row-major) + b_ih
// One wave computes a 16(M) x 64(N) strip: 4 accumulators, A-frag reused 4x.
// Waves total: (16384/16) * (2048/64) = 32768 -> 4096 blocks x 8 waves.
// ---------------------------------------------------------------------------
__global__ void __launch_bounds__(256)
xgate_gemm_kernel(const float* __restrict__ X,     // [T*B, H]
                  const float* __restrict__ Wih,   // [2H, H]
                  const float* __restrict__ bih,   // [2H]
                  float* __restrict__ XG)          // [T*B, 2H]
{
    const int lane  = threadIdx.x & 31;
    const int wave  = blockIdx.x * (blockDim.x >> 5) + (threadIdx.x >> 5);
    const int NGRP  = G2 / 64;                  // 32 column groups
    const int mTile = wave / NGRP;
    const int nGrp  = wave % NGRP;
    const int m0    = mTile * 16;
    const int n0    = nGrp * 64;
    const int hi    = lane >> 4;                // 0/1 selects K-pair
    const int lo    = lane & 15;                // row (A) / col (B)

    v8f acc[4] = {v8f{}, v8f{}, v8f{}, v8f{}};

    const float* arow = X + (size_t)(m0 + lo) * HH + 2 * hi;
    const float* b0   = Wih + (size_t)(n0 +  0 + lo) * HH + 2 * hi;
    const float* b1   = Wih + (size_t)(n0 + 16 + lo) * HH + 2 * hi;
    const float* b2   = Wih + (size_t)(n0 + 32 + lo) * HH + 2 * hi;
    const float* b3   = Wih + (size_t)(n0 + 48 + lo) * HH + 2 * hi;

    #pragma unroll 4
    for (int k = 0; k < HH; k += 4) {
        v2f a = *(const v2f*)(arow + k);
        acc[0] = wmma_f32(a, *(const v2f*)(b0 + k), acc[0]);
        acc[1] = wmma_f32(a, *(const v2f*)(b1 + k), acc[1]);
        acc[2] = wmma_f32(a, *(const v2f*)(b2 + k), acc[2]);
        acc[3] = wmma_f32(a, *(const v2f*)(b3 + k), acc[3]);
    }

    #pragma unroll
    for (int j = 0; j < 4; ++j) {
        const int n = n0 + j * 16 + lo;
        const float bias = bih[n];
        #pragma unroll
        for (int r = 0; r < 8; ++r) {
            const int m = m0 + r + 8 * hi;
            XG[(size_t)m * G2 + n] = acc[j][r] + bias;
        }
    }
}

// ---------------------------------------------------------------------------
// Phase 2: one recurrence step. Each wave owns one (m_tile, h_tile):
//   accf = c @ Whh^T cols [h0, h0+16)        (f-gate)
//   acco = c @ Whh^T cols [H+h0, H+h0+16)    (o-gate)
// FUSED variant additionally accumulates x_t @ Wih^T (no xg workspace).
// 2 m-tiles * 64 h-tiles = 128 waves -> 16 blocks x 256 threads.
// ---------------------------------------------------------------------------
template <bool FUSED>
__global__ void __launch_bounds__(256)
regu_step_kernel(const float* __restrict__ x_t,   // [B, H]
                 const float* __restrict__ xg_t,  // [B, 2H] (unused if FUSED)
                 const float* __restrict__ c_in,  // [B, H]
                 float* __restrict__ c_out,       // [B, H]
                 const float* __restrict__ Whh,   // [2H, H]
                 const float* __restrict__ Wih,   // [2H, H] (FUSED only)
                 const float* __restrict__ bhh,   // [2H]
                 const float* __restrict__ bih,   // [2H]   (FUSED only)
                 float* __restrict__ out_t)       // [B, H]
{
    const int lane  = threadIdx.x & 31;
    const int wave  = blockIdx.x * (blockDim.x >> 5) + (threadIdx.x >> 5);
    const int mTile = wave & 1;        // B=32 -> 2 tiles of 16
    const int hTile = wave >> 1;       // 64 tiles of 16 cols
    const int m0    = mTile * 16;
    const int h0    = hTile * 16;
    const int hi    = lane >> 4;
    const int lo    = lane & 15;

    v8f accf = v8f{};
    v8f acco = v8f{};

    const float* arowc = c_in + (size_t)(m0 + lo) * HH + 2 * hi;
    const float* arowx = x_t + (size_t)(m0 + lo) * HH + 2 * hi;
    const float* bfh   = Whh + (size_t)(h0 + lo) * HH + 2 * hi;        // f-gate rows
    const float* boh   = Whh + (size_t)(HH + h0 + lo) * HH + 2 * hi;   // o-gate rows
    const float* bfi   = FUSED ? (Wih + (size_t)(h0 + lo) * HH + 2 * hi) : nullptr;
    const float* boi   = FUSED ? (Wih + (size_t)(HH + h0 + lo) * HH + 2 * hi) : nullptr;

    #pragma unroll 4
    for (int k = 0; k < HH; k += 4) {
        v2f ac = *(const v2f*)(arowc + k);
        accf = wmma_f32(ac, *(const v2f*)(bfh + k), accf);
        acco = wmma_f32(ac, *(const v2f*)(boh + k), acco);
        if (FUSED) {
            v2f ax = *(const v2f*)(arowx + k);
            accf = wmma_f32(ax, *(const v2f*)(bfi + k), accf);
            acco = wmma_f32(ax, *(const v2f*)(boi + k), acco);
        }
    }

    float biasF = bhh[h0 + lo];
    float biasO = bhh[HH + h0 + lo];
    if (FUSED) {
        biasF += bih[h0 + lo];
        biasO += bih[HH + h0 + lo];
    }

    #pragma unroll
    for (int r = 0; r < 8; ++r) {
        const int m = m0 + r + 8 * hi;     // batch row
        const int h = h0 + lo;             // hidden col
        float gf = accf[r] + biasF;
        float go = acco[r] + biasO;
        if (!FUSED) {
            gf += xg_t[(size_t)m * G2 + h];
            go += xg_t[(size_t)m * G2 + HH + h];
        }
        const float f  = sigmoidf_(gf);
        const float o  = sigmoidf_(go);
        const float cv = c_in[(size_t)m * HH + h];
        const float xv = x_t[(size_t)m * HH + h];
        const float cy = (1.0f - f) * cv + f * tanhf(xv);
        const float hy = (1.0f - o) * xv + o * tanhf(cy);
        c_out[(size_t)m * HH + h] = cy;
        out_t[(size_t)m * HH + h] = hy;
    }
}

__global__ void copy_state_kernel(const float* __restrict__ src,
                                  float* __restrict__ dst, int n) {
    int i = blockIdx.x * blockDim.x + threadIdx.x;
    if (i < n) dst[i] = src[i];
}

extern "C" void kernel_launch(void* const* d_in, const int* in_sizes, int n_in,
                              void* d_out, int out_size, void* d_ws, size_t ws_size,
                              hipStream_t stream) {
    const float* inputs = (const float*)d_in[0];  // [T, B, H]
    const float* state  = (const float*)d_in[1];  // [B, H]
    const float* w_ih   = (const float*)d_in[2];  // [2H, H]
    const float* w_hh   = (const float*)d_in[3];  // [2H, H]
    const float* b_ih   = (const float*)d_in[4];  // [2H]
    const float* b_hh   = (const float*)d_in[5];  // [2H]

    float* out   = (float*)d_out;                 // [T*B*H] outputs ++ [B*H] c_final
    float* cbuf0 = out + (size_t)TT * BB * HH;    // c ping buffer == c_final slot
    float* cbuf1 = (float*)d_ws;                  // c pong buffer (128 KB)

    const size_t xg_off = (((size_t)BB * HH * sizeof(float)) + 255) & ~(size_t)255;
    float* xg = (float*)((char*)d_ws + xg_off);
    const size_t need = xg_off + (size_t)TT * BB * G2 * sizeof(float);  // ~128 MB
    const bool hoist = (ws_size >= need);

    // c0 = state
    copy_state_kernel<<<(BB * HH + 255) / 256, 256, 0, stream>>>(state, cbuf0, BB * HH);

    if (hoist) {
        // 32768 waves = 4096 blocks of 8 waves
        xgate_gemm_kernel<<<4096, 256, 0, stream>>>(inputs, w_ih, b_ih, xg);
    }

    for (int t = 0; t < TT; ++t) {
        float* cin  = (t & 1) ? cbuf1 : cbuf0;
        float* cout = (t & 1) ? cbuf0 : cbuf1;
        const float* x_t = inputs + (size_t)t * BB * HH;
        float* o_t = out + (size_t)t * BB * HH;
        if (hoist) {
            regu_step_kernel<false><<<16, 256, 0, stream>>>(
                x_t, xg + (size_t)t * BB * G2, cin, cout, w_hh, nullptr, b_hh, nullptr, o_t);
        } else {
            regu_step_kernel<true><<<16, 256, 0, stream>>>(
                x_t, nullptr, cin, cout, w_hh, w_ih, b_hh, b_ih, o_t);
        }
    }
    // t=511 is odd -> cout == cbuf0 == d_out tail: c_final in place.
}